// GAT_26817775796801
// MI455X (gfx1250) — compile-verified
//
#include <hip/hip_runtime.h>
#include <hip/hip_bf16.h>

typedef __attribute__((ext_vector_type(2))) float v2f;
typedef __attribute__((ext_vector_type(4))) float v4f;
typedef __attribute__((ext_vector_type(8))) float v8f;

#define NEG_SLOPE 0.01f

// =====================================================================
// GEMM C[N,COLS] = A[N,K] @ B[K,COLS] using V_WMMA_F32_16X16X4_F32.
// One wave computes a 16 x COLS strip. Wave-uniform guard keeps EXEC full.
//
// A-frag (16x4 f32, 2 VGPRs): lane l holds row (l&15), K = (l>>4)*2 + {0,1}
// B-frag (4x16 f32, 2 VGPRs): lane l holds col (l&15), K = (l>>4)*2 + {0,1}
// C/D   (16x16 f32, 8 VGPRs): VGPR r <-> row r + (l>>4)*8, col (l&15)
// =====================================================================
template <int K, int COLS>
__global__ __launch_bounds__(256) void gemm_wmma_f32(
    const float* __restrict__ A, const float* __restrict__ B,
    float* __restrict__ C, int N)
{
  constexpr int NT = COLS / 16;
  const int wid  = (blockIdx.x * blockDim.x + threadIdx.x) >> 5;
  const int lane = threadIdx.x & 31;
  const int row0 = wid * 16;
  if (row0 >= N) return;                       // wave-uniform: EXEC stays full
  const int rA  = row0 + (lane & 15);
  const int kh  = (lane >> 4) << 1;            // 0 or 2
  const int col = lane & 15;

  v8f acc[NT] = {};
  const float* __restrict__ arow = A + (size_t)rA * K + kh;

  for (int k = 0; k < K; k += 4) {
    v2f a = *(const v2f*)(arow + k);           // 8B aligned: even float offset
#pragma unroll
    for (int t = 0; t < NT; ++t) {
      const float* bp = B + (size_t)(k + kh) * COLS + t * 16 + col;
      v2f b;
      b.x = bp[0];
      b.y = bp[COLS];
      acc[t] = __builtin_amdgcn_wmma_f32_16x16x4_f32(
          false, a, false, b, (short)0, acc[t], false, false);
    }
  }

  const int rBase = row0 + ((lane >> 4) << 3);
#pragma unroll
  for (int t = 0; t < NT; ++t) {
#pragma unroll
    for (int r = 0; r < 8; ++r) {
      C[(size_t)(rBase + r) * COLS + t * 16 + col] = acc[t][r];
    }
  }
}

// =====================================================================
// Per-(node, head) attention scores: s1 = <h, a1[k]>, s2 = <h, a2[k]>.
// H is [N, 8*F] (8 heads of F features). One thread per (n, k).
// =====================================================================
template <int F>
__global__ __launch_bounds__(256) void gat_scores(
    const float* __restrict__ H, const float* __restrict__ A1,
    const float* __restrict__ A2, float* __restrict__ S1,
    float* __restrict__ S2, int N)
{
  const int i = blockIdx.x * blockDim.x + threadIdx.x;   // i = n*8 + k
  if (i >= N * 8) return;
  const int k = i & 7;
  const float* h = H + (size_t)(i >> 3) * (8 * F) + k * F;
  float d1 = 0.f, d2 = 0.f;
#pragma unroll
  for (int f = 0; f < F; ++f) {
    const float v = h[f];
    d1 = fmaf(v, A1[k * F + f], d1);
    d2 = fmaf(v, A2[k * F + f], d2);
  }
  S1[i] = d1;
  S2[i] = d2;
}

// =====================================================================
// Layer-1 aggregation + ELU. One wave per node; lane d <-> neighbor d.
// 8 heads x 8 features. Softmax over the 32 lanes via shfl_xor butterfly.
// =====================================================================
__global__ __launch_bounds__(256) void gat_agg1_elu(
    const float* __restrict__ H,   // [N,64] layer-1 features
    const float* __restrict__ S1,  // [N,8]
    const float* __restrict__ S2,  // [N,8]
    const int*   __restrict__ NB,  // [N,32]
    float* __restrict__ X2, int N) // [N,64] output (ELU applied)
{
  const int n    = (blockIdx.x * blockDim.x + threadIdx.x) >> 5;
  const int lane = threadIdx.x & 31;
  if (n >= N) return;                          // wave-uniform
  const int nb = NB[(size_t)n * 32 + lane];

  float e[8];
  {
    const v4f s2a = *(const v4f*)(S2 + (size_t)nb * 8);
    const v4f s2b = *(const v4f*)(S2 + (size_t)nb * 8 + 4);
    const float s2v[8] = {s2a.x, s2a.y, s2a.z, s2a.w,
                          s2b.x, s2b.y, s2b.z, s2b.w};
#pragma unroll
    for (int k = 0; k < 8; ++k) {
      const float v = S1[(size_t)n * 8 + k] + s2v[k];
      e[k] = v > 0.f ? v : NEG_SLOPE * v;
    }
  }

  float alpha[8];
#pragma unroll
  for (int k = 0; k < 8; ++k) {
    float m = e[k];
#pragma unroll
    for (int off = 16; off > 0; off >>= 1)
      m = fmaxf(m, __shfl_xor(m, off, 32));
    const float p = __expf(e[k] - m);
    float s = p;
#pragma unroll
    for (int off = 16; off > 0; off >>= 1)
      s += __shfl_xor(s, off, 32);
    alpha[k] = p / s;
  }

#pragma unroll
  for (int k = 0; k < 8; ++k) {
    const v4f ha = *(const v4f*)(H + (size_t)nb * 64 + k * 8);
    const v4f hb = *(const v4f*)(H + (size_t)nb * 64 + k * 8 + 4);
    float t[8] = {ha.x, ha.y, ha.z, ha.w, hb.x, hb.y, hb.z, hb.w};
#pragma unroll
    for (int f = 0; f < 8; ++f) t[f] *= alpha[k];
#pragma unroll
    for (int off = 16; off > 0; off >>= 1) {
#pragma unroll
      for (int f = 0; f < 8; ++f) t[f] += __shfl_xor(t[f], off, 32);
    }
    if (lane == 0) {
#pragma unroll
      for (int f = 0; f < 8; ++f) {
        const float v = t[f];
        t[f] = v > 0.f ? v : (__expf(v) - 1.0f);   // ELU(alpha=1)
      }
      const v4f o0 = {t[0], t[1], t[2], t[3]};
      const v4f o1 = {t[4], t[5], t[6], t[7]};
      *(v4f*)(X2 + (size_t)n * 64 + k * 8)     = o0;
      *(v4f*)(X2 + (size_t)n * 64 + k * 8 + 4) = o1;
    }
  }
}

// =====================================================================
// Layer-2 aggregation + mean over heads + class softmax. Wave per node.
// 8 heads x 16 features/classes.
// =====================================================================
__global__ __launch_bounds__(256) void gat_agg2_out(
    const float* __restrict__ H,    // [N,128] layer-2 features
    const float* __restrict__ S1,   // [N,8]
    const float* __restrict__ S2,   // [N,8]
    const int*   __restrict__ NB,   // [N,32]
    float* __restrict__ OUT, int N) // [N,16]
{
  const int n    = (blockIdx.x * blockDim.x + threadIdx.x) >> 5;
  const int lane = threadIdx.x & 31;
  if (n >= N) return;                          // wave-uniform
  const int nb = NB[(size_t)n * 32 + lane];

  float e[8];
  {
    const v4f s2a = *(const v4f*)(S2 + (size_t)nb * 8);
    const v4f s2b = *(const v4f*)(S2 + (size_t)nb * 8 + 4);
    const float s2v[8] = {s2a.x, s2a.y, s2a.z, s2a.w,
                          s2b.x, s2b.y, s2b.z, s2b.w};
#pragma unroll
    for (int k = 0; k < 8; ++k) {
      const float v = S1[(size_t)n * 8 + k] + s2v[k];
      e[k] = v > 0.f ? v : NEG_SLOPE * v;
    }
  }

  float alpha[8];
#pragma unroll
  for (int k = 0; k < 8; ++k) {
    float m = e[k];
#pragma unroll
    for (int off = 16; off > 0; off >>= 1)
      m = fmaxf(m, __shfl_xor(m, off, 32));
    const float p = __expf(e[k] - m);
    float s = p;
#pragma unroll
    for (int off = 16; off > 0; off >>= 1)
      s += __shfl_xor(s, off, 32);
    alpha[k] = p / s;
  }

  float cls[16] = {};
#pragma unroll
  for (int k = 0; k < 8; ++k) {
    float t[16];
#pragma unroll
    for (int q = 0; q < 4; ++q) {
      const v4f h4 = *(const v4f*)(H + (size_t)nb * 128 + k * 16 + q * 4);
      t[q * 4 + 0] = h4.x; t[q * 4 + 1] = h4.y;
      t[q * 4 + 2] = h4.z; t[q * 4 + 3] = h4.w;
    }
    const float a = alpha[k];
#pragma unroll
    for (int f = 0; f < 16; ++f) t[f] *= a;
#pragma unroll
    for (int off = 16; off > 0; off >>= 1) {
#pragma unroll
      for (int f = 0; f < 16; ++f) t[f] += __shfl_xor(t[f], off, 32);
    }
#pragma unroll
    for (int f = 0; f < 16; ++f) cls[f] += t[f];
  }

  // mean over 8 heads
#pragma unroll
  for (int f = 0; f < 16; ++f) cls[f] *= 0.125f;
  // softmax over 16 classes (every lane computes it; lane 0 stores)
  float m = cls[0];
#pragma unroll
  for (int f = 1; f < 16; ++f) m = fmaxf(m, cls[f]);
  float s = 0.f;
#pragma unroll
  for (int f = 0; f < 16; ++f) { cls[f] = __expf(cls[f] - m); s += cls[f]; }
  const float inv = 1.f / s;

  if (lane == 0) {
#pragma unroll
    for (int q = 0; q < 4; ++q) {
      const v4f o = {cls[q * 4 + 0] * inv, cls[q * 4 + 1] * inv,
                     cls[q * 4 + 2] * inv, cls[q * 4 + 3] * inv};
      *(v4f*)(OUT + (size_t)n * 16 + q * 4) = o;
    }
  }
}

// =====================================================================
// Launch
// =====================================================================
extern "C" void kernel_launch(void* const* d_in, const int* in_sizes, int n_in,
                              void* d_out, int out_size, void* d_ws, size_t ws_size,
                              hipStream_t stream) {
  const float* X   = (const float*)d_in[0];  // [N,128]
  const int*   NB  = (const int*)  d_in[1];  // [N,32]
  const float* W1  = (const float*)d_in[2];  // [128,64]
  const float* A11 = (const float*)d_in[3];  // [8,8]
  const float* A21 = (const float*)d_in[4];  // [8,8]
  const float* W2  = (const float*)d_in[5];  // [64,128]
  const float* A12 = (const float*)d_in[6];  // [8,16]
  const float* A22 = (const float*)d_in[7];  // [8,16]
  float* OUT = (float*)d_out;                // [N,16]

  const int N = in_sizes[0] / 128;

  // Workspace layout (floats), with h1/h2 time-aliased at offset 0:
  //   [0,          N*128)  : h1 (layer1 feats, N*64 used) then h2 (N*128)
  //   [N*128,      N*136)  : s1 (reused across layers)
  //   [N*136,      N*144)  : s2 (reused across layers)
  //   [N*144,      N*208)  : x2 (ELU'd layer-1 output, N*64)
  float* ws = (float*)d_ws;
  float* hbuf = ws;                          // h1 then h2
  float* s1   = ws + (size_t)N * 128;
  float* s2   = s1 + (size_t)N * 8;
  float* x2   = s2 + (size_t)N * 8;

  const int threadsPerBlock = 256;

  // ---- Layer 1 ----
  {
    const int waves  = (N + 15) / 16;
    const int blocks = (waves * 32 + threadsPerBlock - 1) / threadsPerBlock;
    gemm_wmma_f32<128, 64><<<blocks, threadsPerBlock, 0, stream>>>(X, W1, hbuf, N);
  }
  {
    const int blocks = (N * 8 + threadsPerBlock - 1) / threadsPerBlock;
    gat_scores<8><<<blocks, threadsPerBlock, 0, stream>>>(hbuf, A11, A21, s1, s2, N);
  }
  {
    const int blocks = (N * 32 + threadsPerBlock - 1) / threadsPerBlock;
    gat_agg1_elu<<<blocks, threadsPerBlock, 0, stream>>>(hbuf, s1, s2, NB, x2, N);
  }

  // ---- Layer 2 ----
  {
    const int waves  = (N + 15) / 16;
    const int blocks = (waves * 32 + threadsPerBlock - 1) / threadsPerBlock;
    gemm_wmma_f32<64, 128><<<blocks, threadsPerBlock, 0, stream>>>(x2, W2, hbuf, N);
  }
  {
    const int blocks = (N * 8 + threadsPerBlock - 1) / threadsPerBlock;
    gat_scores<16><<<blocks, threadsPerBlock, 0, stream>>>(hbuf, A12, A22, s1, s2, N);
  }
  {
    const int blocks = (N * 32 + threadsPerBlock - 1) / threadsPerBlock;
    gat_agg2_out<<<blocks, threadsPerBlock, 0, stream>>>(hbuf, s1, s2, NB, OUT, N);
  }
}